// FNetEncoderBlock_29549374997093
// MI455X (gfx1250) — compile-verified
//
#include <hip/hip_runtime.h>
#include <hip/hip_bf16.h>
#include <math.h>

// FNet encoder block for MI455X (gfx1250, wave32, WMMA f32).
//   B=4, S=4096, D=768, M=3072, rows R = B*S = 16384.
//
// Pipeline:
//   K0 init_dft      : C[d][j]=cos(2*pi*d*j/D), Sm[d][j]=-sin(...)
//   K1 gemm (x2)     : yr = X*C, yi = X*Sm          (WMMA f32 16x16x4)
//   K2 fft_seq       : radix-2 FFT len 4096 along seq in LDS; keep Re -> yr
//   K3 add_ln        : xln = LN(x + yr)
//   K4 gemm          : h  = GELU(xln*W1^T + b1)
//   K5 gemm          : ff = h*W2^T + b2             (into yi buffer)
//   K6 add_ln        : out = LN(xln + ff)
//
// GEMM: block tile 256x64, BK=32, 8 waves x (64x32 per wave, 4x2 WMMA frags).
// LDS tiles stored k-contiguous (row stride 36 words) so every WMMA fragment
// is one aligned ds_load_b64, conflict-free across the 16-lane half-wave.
//
// Workspace (floats): dft_re[768^2] dft_im[768^2] yr[R*D] yi[R*D] xln[R*D] h[R*M]

typedef __attribute__((ext_vector_type(2))) float v2f;
typedef __attribute__((ext_vector_type(8))) float v8f;

#define TWO_PI 6.28318530717958647692f

static __device__ __forceinline__ v8f wmma4(v2f a, v2f b, v8f c) {
  // V_WMMA_F32_16X16X4_F32 : D = A(16x4) * B(4x16) + C(16x16), all f32.
  return __builtin_amdgcn_wmma_f32_16x16x4_f32(false, a, false, b, (short)0, c, false, false);
}

// ---------------------------------------------------------------------------
__global__ __launch_bounds__(256) void init_dft_kernel(float* __restrict__ dre,
                                                       float* __restrict__ dim_) {
  int id = blockIdx.x * 256 + threadIdx.x;
  if (id >= 768 * 768) return;
  int d = id / 768;
  int j = id - d * 768;
  int p = (d * j) % 768;                       // exact reduction, fits int32
  float ang = -(TWO_PI / 768.0f) * (float)p;   // angle of e^{-2*pi*i*d*j/768}
  dre[id] = cosf(ang);
  dim_[id] = sinf(ang);                        // = -sin(2*pi*d*j/768)
}

// ---------------------------------------------------------------------------
// WMMA f32 GEMM:  C[r][n] = sum_k A[r][k] * Bw[k][n]  (+ epilogue)
//   BT=false : Bw[k][n] = Bsrc[k*ldb + n]   (DFT matrices, row-major KxN)
//   BT=true  : Bw[k][n] = Bsrc[n*ldb + k]   (W1/W2: out = A * W^T)
//   EPI: 0 none, 1 +bias, 2 +bias then exact GELU
// ---------------------------------------------------------------------------
#define BKP 36  // BK=32 + 4 pad: keeps ds_store_b128 16B-aligned, b64 reads conflict-free

template <int EPI, bool BT>
__global__ __launch_bounds__(256) void gemm_f32_kernel(
    const float* __restrict__ A, const float* __restrict__ Bsrc,
    const float* __restrict__ bias, float* __restrict__ Cout,
    int K, int lda, int ldb, int ldc) {
  __shared__ float As[256][BKP];  // [m][k] : k-contiguous
  __shared__ float Bs[64][BKP];   // [n][k] : k-contiguous

  const int tid = threadIdx.x;
  const int lane = tid & 31;
  const int wave = tid >> 5;
  const int l = lane & 15;        // row/col within a 16-frag
  const int half = lane >> 4;     // K sub-pair {0,1} vs {2,3}; M sub-half on store
  const int row0 = (wave & 3) * 64;   // 4 waves along M, 64 rows each
  const int col0 = (wave >> 2) * 32;  // 2 waves along N, 32 cols each
  const int rowBase = blockIdx.y * 256;
  const int colBase = blockIdx.x * 64;

  v8f acc[4][2];
#pragma unroll
  for (int i = 0; i < 4; ++i)
#pragma unroll
    for (int j = 0; j < 2; ++j) acc[i][j] = (v8f){0.f, 0.f, 0.f, 0.f, 0.f, 0.f, 0.f, 0.f};

  for (int kb = 0; kb < K; kb += 32) {
    __syncthreads();
    // ---- stage A tile: 256 rows x 32 k. Memory is k-contiguous -> direct b128 copy.
    {
      const int r_sub = tid >> 3;        // 0..31
      const int kq = (tid & 7) * 4;      // 0,4,...,28
#pragma unroll
      for (int i = 0; i < 8; ++i) {
        int r = r_sub + i * 32;
        float4 w = *reinterpret_cast<const float4*>(
            &A[(size_t)(rowBase + r) * lda + kb + kq]);
        *reinterpret_cast<float4*>(&As[r][kq]) = w;
      }
      if (kb + 32 < K)  // hint next K panel (global_prefetch_b8)
        __builtin_prefetch(&A[(size_t)(rowBase + r_sub) * lda + kb + 32], 0, 1);
    }
    // ---- stage B tile: 64 n x 32 k
    if (BT) {  // Bsrc row-major [n][k] -> direct b128 copy
      const int n_sub = tid >> 3;
      const int kq = (tid & 7) * 4;
#pragma unroll
      for (int i = 0; i < 2; ++i) {
        int n = n_sub + i * 32;
        float4 w = *reinterpret_cast<const float4*>(
            &Bsrc[(size_t)(colBase + n) * ldb + kb + kq]);
        *reinterpret_cast<float4*>(&Bs[n][kq]) = w;
      }
    } else {  // Bsrc row-major [k][n] -> transpose while staging
      const int k_sub = tid >> 4;        // 0..15
      const int nq = (tid & 15) * 4;     // 0,4,...,60
#pragma unroll
      for (int i = 0; i < 2; ++i) {
        int k = k_sub + i * 16;
        float4 w = *reinterpret_cast<const float4*>(
            &Bsrc[(size_t)(kb + k) * ldb + colBase + nq]);
        Bs[nq + 0][k] = w.x;
        Bs[nq + 1][k] = w.y;
        Bs[nq + 2][k] = w.z;
        Bs[nq + 3][k] = w.w;
      }
    }
    __syncthreads();

    // ---- 8 WMMA k-steps of 4; every fragment is one ds_load_b64
#pragma unroll
    for (int kk = 0; kk < 32; kk += 4) {
      const int k0 = kk + half * 2;  // lanes 0-15 hold K={0,1}, lanes 16-31 K={2,3}
      v2f a[4], b[2];
#pragma unroll
      for (int am = 0; am < 4; ++am)
        a[am] = *reinterpret_cast<const v2f*>(&As[row0 + am * 16 + l][k0]);
#pragma unroll
      for (int bn = 0; bn < 2; ++bn)
        b[bn] = *reinterpret_cast<const v2f*>(&Bs[col0 + bn * 16 + l][k0]);
#pragma unroll
      for (int am = 0; am < 4; ++am)
#pragma unroll
        for (int bn = 0; bn < 2; ++bn) acc[am][bn] = wmma4(a[am], b[bn], acc[am][bn]);
    }
  }

  // ---- epilogue + store (C/D layout: VGPR g -> M = g + 8*half, lane l -> N)
#pragma unroll
  for (int am = 0; am < 4; ++am)
#pragma unroll
    for (int bn = 0; bn < 2; ++bn) {
      const int c = colBase + col0 + bn * 16 + l;
      float bv = 0.0f;
      if constexpr (EPI != 0) bv = bias[c];
#pragma unroll
      for (int g = 0; g < 8; ++g) {
        int r = rowBase + row0 + am * 16 + half * 8 + g;
        float v = acc[am][bn][g] + bv;
        if constexpr (EPI == 2)
          v = 0.5f * v * (1.0f + erff(v * 0.70710678118654752f));  // exact GELU
        Cout[(size_t)r * ldc + c] = v;
      }
    }
}

// ---------------------------------------------------------------------------
// K2: complex radix-2 FFT (length 4096) along seq for one (b,d) column.
// ---------------------------------------------------------------------------
__global__ __launch_bounds__(256) void fft_seq_kernel(float* __restrict__ yr,
                                                      const float* __restrict__ yi) {
  __shared__ float2 sh[4096];
  const int d = blockIdx.x;   // 0..767
  const int b = blockIdx.y;   // 0..3
  const size_t base = ((size_t)b * 4096) * 768 + d;

  for (int s = threadIdx.x; s < 4096; s += 256) {
    int j = (int)(__brev((unsigned)s) >> 20);  // 12-bit bit-reverse
    size_t idx = base + (size_t)s * 768;
    sh[j] = make_float2(yr[idx], yi[idx]);
  }
  __syncthreads();

  for (int half = 1; half < 4096; half <<= 1) {
    const int m = half << 1;
    const float wstep = -TWO_PI / (float)m;
    for (int t = threadIdx.x; t < 2048; t += 256) {
      const int pos = t & (half - 1);
      const int i1 = (t << 1) - pos;
      const int i2 = i1 + half;
      float sn, cs;
      __sincosf(wstep * (float)pos, &sn, &cs);
      float2 u = sh[i1];
      float2 w = sh[i2];
      float tr = w.x * cs - w.y * sn;
      float ti = w.x * sn + w.y * cs;
      sh[i2] = make_float2(u.x - tr, u.y - ti);
      sh[i1] = make_float2(u.x + tr, u.y + ti);
    }
    __syncthreads();
  }

  for (int s = threadIdx.x; s < 4096; s += 256)
    yr[base + (size_t)s * 768] = sh[s].x;  // keep real part only
}

// ---------------------------------------------------------------------------
// K3/K6: out = LayerNorm(a + b) over last dim (768), two-pass.
// ---------------------------------------------------------------------------
__global__ __launch_bounds__(256) void add_ln_kernel(const float* __restrict__ a,
                                                     const float* __restrict__ b,
                                                     float* __restrict__ out) {
  __shared__ float red[256];
  const int tid = threadIdx.x;
  const size_t base = (size_t)blockIdx.x * 768;

  float v[3];
  float s = 0.0f;
#pragma unroll
  for (int i = 0; i < 3; ++i) {
    int e = tid + i * 256;
    v[i] = a[base + e] + b[base + e];
    s += v[i];
  }
  red[tid] = s;
  __syncthreads();
  for (int off = 128; off > 0; off >>= 1) {
    if (tid < off) red[tid] += red[tid + off];
    __syncthreads();
  }
  const float mean = red[0] * (1.0f / 768.0f);
  __syncthreads();

  float q = 0.0f;
#pragma unroll
  for (int i = 0; i < 3; ++i) {
    float dlt = v[i] - mean;
    q += dlt * dlt;
  }
  red[tid] = q;
  __syncthreads();
  for (int off = 128; off > 0; off >>= 1) {
    if (tid < off) red[tid] += red[tid + off];
    __syncthreads();
  }
  const float rstd = rsqrtf(red[0] * (1.0f / 768.0f) + 1e-5f);
#pragma unroll
  for (int i = 0; i < 3; ++i) {
    int e = tid + i * 256;
    out[base + e] = (v[i] - mean) * rstd;
  }
}

// ---------------------------------------------------------------------------
extern "C" void kernel_launch(void* const* d_in, const int* in_sizes, int n_in,
                              void* d_out, int out_size, void* d_ws, size_t ws_size,
                              hipStream_t stream) {
  (void)in_sizes; (void)n_in; (void)out_size; (void)ws_size;

  constexpr int S = 4096, D = 768, M = 3072, R = 4 * S;  // 16384 token rows

  const float* x  = (const float*)d_in[0];  // (B,S,D)
  const float* W1 = (const float*)d_in[1];  // (M,D)
  const float* b1 = (const float*)d_in[2];  // (M)
  const float* W2 = (const float*)d_in[3];  // (D,M)
  const float* b2 = (const float*)d_in[4];  // (D)
  float* out = (float*)d_out;               // (B,S,D)

  float* ws = (float*)d_ws;
  float* dft_re = ws;                       // 768*768
  float* dft_im = dft_re + (size_t)D * D;   // 768*768
  float* yr  = dft_im + (size_t)D * D;      // R*D
  float* yi  = yr + (size_t)R * D;          // R*D  (later reused for ff)
  float* xln = yi + (size_t)R * D;          // R*D
  float* h   = xln + (size_t)R * D;         // R*M

  // K0: DFT matrices
  init_dft_kernel<<<(D * D + 255) / 256, 256, 0, stream>>>(dft_re, dft_im);

  // K1: feature-axis DFT, real & imag parts (X is (R x D) row-major)
  gemm_f32_kernel<0, false><<<dim3(D / 64, R / 256), 256, 0, stream>>>(
      x, dft_re, nullptr, yr, /*K=*/D, /*lda=*/D, /*ldb=*/D, /*ldc=*/D);
  gemm_f32_kernel<0, false><<<dim3(D / 64, R / 256), 256, 0, stream>>>(
      x, dft_im, nullptr, yi, D, D, D, D);

  // K2: seq-axis FFT per (b,d) column; real part -> yr (the mixed signal)
  fft_seq_kernel<<<dim3(D, 4), 256, 0, stream>>>(yr, yi);

  // K3: xln = LN(x + mixed)
  add_ln_kernel<<<R, 256, 0, stream>>>(x, yr, xln);

  // K4: h = GELU(xln * W1^T + b1)   (R x M)
  gemm_f32_kernel<2, true><<<dim3(M / 64, R / 256), 256, 0, stream>>>(
      xln, W1, b1, h, /*K=*/D, /*lda=*/D, /*ldb=*/D, /*ldc=*/M);

  // K5: ff = h * W2^T + b2          (R x D), stored into yi
  gemm_f32_kernel<1, true><<<dim3(D / 64, R / 256), 256, 0, stream>>>(
      h, W2, b2, yi, /*K=*/M, /*lda=*/M, /*ldb=*/M, /*ldc=*/D);

  // K6: out = LN(xln + ff)
  add_ln_kernel<<<R, 256, 0, stream>>>(xln, yi, out);
}